// ProposalToDetectBox_26491358282130
// MI455X (gfx1250) — compile-verified
//
#include <hip/hip_runtime.h>
#include <stdint.h>

// Problem constants (from reference setup_inputs)
constexpr int kB = 8;
constexpr int kN = 2000;
constexpr int kC = 81;
constexpr int kMaxOut = 300;
constexpr float kScoreT = 0.05f;
constexpr float kIouT = 0.5f;

typedef __attribute__((ext_vector_type(4))) unsigned int u32x4;
typedef __attribute__((ext_vector_type(8))) int i32x8;
typedef __attribute__((ext_vector_type(4))) int i32x4;

// ---------------------------------------------------------------------------
// Kernel 1: softmax -> fg score/class, delta gather, box regression.
// One thread per (b,n). Inputs are L2-resident (26MB << 192MB L2).
// ---------------------------------------------------------------------------
__global__ __launch_bounds__(256) void prep_kernel(
    const float* __restrict__ deltas,      // (B,N,C,4)
    const float* __restrict__ logits,      // (B,N,C)
    const float* __restrict__ proposals,   // (B,N,5)
    float* __restrict__ ws_boxes,          // (B*N,4)
    float* __restrict__ ws_live)           // (B*N)
{
  int t = blockIdx.x * blockDim.x + threadIdx.x;
  if (t >= kB * kN) return;

  const float* lg = logits + (size_t)t * kC;

  // max over all classes (softmax stabilizer) + argmax over classes 1..80
  float m_all = lg[0];
  float m_fg = -3.402823466e38f;
  int cls = 1;
  for (int c = 1; c < kC; ++c) {
    float v = lg[c];
    m_all = fmaxf(m_all, v);
    if (v > m_fg) { m_fg = v; cls = c; }   // strict '>' => first max, like jnp.argmax
  }
  float sum = 0.0f;
  for (int c = 0; c < kC; ++c) sum += expf(lg[c] - m_all);
  float score = expf(m_fg - m_all) / sum;  // == max of softmax over classes 1..80

  // gather delta for fg class (16B-aligned float4)
  const float4 d = *(const float4*)(deltas + ((size_t)t * kC + cls) * 4);
  float dy = d.x * 0.1f, dx = d.y * 0.1f, dh = d.z * 0.2f, dw = d.w * 0.2f;

  const float* pp = proposals + (size_t)t * 5;
  float y1 = pp[0], x1 = pp[1], y2 = pp[2], x2 = pp[3];
  float h = y2 - y1, w = x2 - x1;
  float cy = (y2 + y1) * 0.5f + dy * h;
  float cx = (x2 + x1) * 0.5f + dx * w;
  h *= expf(dh);
  w *= expf(dw);

  float4 ob = { cy - h * 0.5f, cx - w * 0.5f, cy + h * 0.5f, cx + w * 0.5f };
  ((float4*)ws_boxes)[t] = ob;
  ws_live[t] = (score > kScoreT) ? score : -1.0f;
}

// ---------------------------------------------------------------------------
// Kernel 2: per-batch sequential NMS. One block (1024 threads = 32 waves) per
// batch. Box tile staged into LDS via the Tensor Data Mover (CDNA5 TDM);
// per-iteration argmax uses wave32 shuffle tree + one cross-wave LDS pass.
// IoU row recomputed on the fly (never materialize 2000x2000 matrix).
// ---------------------------------------------------------------------------
__global__ __launch_bounds__(1024) void nms_kernel(
    const float* __restrict__ logits,      // (B,N,C)
    const float* __restrict__ ws_boxes,    // (B*N,4)
    const float* __restrict__ ws_live,     // (B*N)
    float* __restrict__ out)               // boxes | scores | logits, flat
{
  const int b = blockIdx.x;
  const int t = threadIdx.x;
  const int lane = t & 31;
  const int wave = t >> 5;

  __shared__ float s_boxes[kN * 4];        // 32000 B
  __shared__ float s_live[2048];           //  8192 B (padded to 2*blockDim)
  __shared__ float s_rv[32];
  __shared__ int   s_ri[32];
  __shared__ float s_box[4];
  __shared__ float s_bval;
  __shared__ int   s_bidx;

  const float* gb = ws_boxes + (size_t)b * kN * 4;

  // --- TDM async load: 8000 fp32 (32KB) global -> LDS, issued by wave 0 ---
  if (t == 0) {
    uint32_t lds_off = (uint32_t)(uintptr_t)s_boxes;   // AS3 offset == low 32 bits
    uint64_t ga = (uint64_t)(uintptr_t)gb;
    // D# group 0: count=1 | lds_addr | global_addr[56:0] | type=2
    u32x4 g0 = { 1u,
                 lds_off,
                 (uint32_t)(ga & 0xFFFFFFFFu),
                 (uint32_t)((ga >> 32) & 0x1FFFFFFu) | (2u << 30) };
    // D# group 1: data_size=4B; 1-D tile: tile_dim0=tensor_dim0=stride=8000
    const int kElems = kN * 4;             // 8000 elements of 4 bytes
    i32x8 g1 = { 0x20000,                  // mask=0, data_size=2 (4B)
                 (int)((kElems & 0xFFFF) << 16), // tensor_dim0 lo16
                 (int)(1u << 16),          // tensor_dim0 hi=0 | tensor_dim1=1
                 (int)(kElems << 16),      // tile_dim0
                 1,                        // tile_dim1=1 | tile_dim2=0
                 kElems,                   // tensor_dim0_stride lo32
                 0, 0 };
    i32x4 gz4 = { 0, 0, 0, 0 };
    i32x8 gz8 = { 0, 0, 0, 0, 0, 0, 0, 0 };
    // amdgpu-toolchain (clang-23) 6-arg form: (g0, g1, g2, g3, extra, cpol)
    __builtin_amdgcn_tensor_load_to_lds(g0, g1, gz4, gz4, gz8, 0);
    __builtin_amdgcn_s_wait_tensorcnt(0); // wave 0 holds the barrier until done
  }

  // live scores -> LDS (pad 2000..2047 with -2 so padding never wins argmax)
  for (int j = t; j < 2048; j += 1024)
    s_live[j] = (j < kN) ? ws_live[(size_t)b * kN + j] : -2.0f;
  __syncthreads();

  // cache this thread's two boxes + areas in registers
  const int j2 = t + 1024;
  float4 bA = *(const float4*)(s_boxes + t * 4);
  float aA = (bA.z - bA.x) * (bA.w - bA.y);
  float4 bB = {0.f, 0.f, 0.f, 0.f};
  float aB = 0.f;
  if (j2 < kN) {
    bB = *(const float4*)(s_boxes + j2 * 4);
    aB = (bB.z - bB.x) * (bB.w - bB.y);
  }

  const float* lgb = logits + (size_t)b * kN * kC;
  float* out_boxes  = out + (size_t)b * kMaxOut * 5;
  float* out_scores = out + (size_t)kB * kMaxOut * 5 + (size_t)b * kMaxOut * 2;
  float* out_logits = out + (size_t)kB * kMaxOut * 7 + (size_t)b * kMaxOut * kC;

  for (int i = 0; i < kMaxOut; ++i) {
    // ---- argmax with first-index tie-break (matches jnp.argmax) ----
    float v0 = s_live[t], v1 = s_live[j2];
    float bv; int bi;
    if (v1 > v0) { bv = v1; bi = j2; } else { bv = v0; bi = t; }
#pragma unroll
    for (int off = 16; off > 0; off >>= 1) {
      float ov = __shfl_down(bv, off);
      int   oi = __shfl_down(bi, off);
      if (ov > bv || (ov == bv && oi < bi)) { bv = ov; bi = oi; }
    }
    if (lane == 0) { s_rv[wave] = bv; s_ri[wave] = bi; }
    __syncthreads();
    if (t < 32) {
      bv = s_rv[t]; bi = s_ri[t];
#pragma unroll
      for (int off = 16; off > 0; off >>= 1) {
        float ov = __shfl_down(bv, off);
        int   oi = __shfl_down(bi, off);
        if (ov > bv || (ov == bv && oi < bi)) { bv = ov; bi = oi; }
      }
      if (t == 0) {
        s_bval = bv;
        s_bidx = bi;
        s_box[0] = s_boxes[bi * 4 + 0];
        s_box[1] = s_boxes[bi * 4 + 1];
        s_box[2] = s_boxes[bi * 4 + 2];
        s_box[3] = s_boxes[bi * 4 + 3];
      }
    }
    __syncthreads();

    const int idx = s_bidx;
    const float mv = s_bval;
    const float flag = (mv > 0.0f) ? 1.0f : 0.0f;   // ok = s[idx] > 0

    // ---- outputs for slot i (disjoint thread ranges; every element written) ----
    if (t < kC) {
      out_logits[(size_t)i * kC + t] = lgb[(size_t)idx * kC + t] * flag;
    } else if (t >= 96 && t < 100) {
      out_boxes[i * 5 + (t - 96)] = s_box[t - 96] * flag;
    } else if (t == 100) {
      out_boxes[i * 5 + 4] = flag;
    } else if (t == 104) {
      out_scores[i * 2 + 0] = (flag > 0.0f) ? mv : 0.0f; // mv == raw fg score when ok
    } else if (t == 105) {
      out_scores[i * 2 + 1] = flag;
    }

    // ---- suppression: IoU(picked, j) > 0.5 -> live[j] = -1; live[idx] = -1 ----
    const float my1 = s_box[0], mx1 = s_box[1], my2 = s_box[2], mx2 = s_box[3];
    const float marea = (my2 - my1) * (mx2 - mx1);
    {
      float yy1 = fmaxf(my1, bA.x), xx1 = fmaxf(mx1, bA.y);
      float yy2 = fminf(my2, bA.z), xx2 = fminf(mx2, bA.w);
      float inter = fmaxf(yy2 - yy1, 0.0f) * fmaxf(xx2 - xx1, 0.0f);
      float iou = inter / (marea + aA - inter + 1e-8f);
      if (iou > kIouT || t == idx) s_live[t] = -1.0f;
    }
    if (j2 < kN) {
      float yy1 = fmaxf(my1, bB.x), xx1 = fmaxf(mx1, bB.y);
      float yy2 = fminf(my2, bB.z), xx2 = fminf(mx2, bB.w);
      float inter = fmaxf(yy2 - yy1, 0.0f) * fmaxf(xx2 - xx1, 0.0f);
      float iou = inter / (marea + aB - inter + 1e-8f);
      if (iou > kIouT || j2 == idx) s_live[j2] = -1.0f;
    }
    __syncthreads();
  }
}

// ---------------------------------------------------------------------------
extern "C" void kernel_launch(void* const* d_in, const int* in_sizes, int n_in,
                              void* d_out, int out_size, void* d_ws, size_t ws_size,
                              hipStream_t stream) {
  const float* deltas    = (const float*)d_in[0];  // (8,2000,81,4)
  const float* logits    = (const float*)d_in[1];  // (8,2000,81)
  const float* proposals = (const float*)d_in[2];  // (8,2000,5)
  float* out = (float*)d_out;

  float* ws_boxes = (float*)d_ws;                  // B*N*4 floats (256 KB)
  float* ws_live  = ws_boxes + (size_t)kB * kN * 4; // B*N floats (64 KB)

  const int total = kB * kN;
  prep_kernel<<<(total + 255) / 256, 256, 0, stream>>>(deltas, logits, proposals,
                                                       ws_boxes, ws_live);
  nms_kernel<<<kB, 1024, 0, stream>>>(logits, ws_boxes, ws_live, out);
}